// CNNCRF_16080357556312
// MI455X (gfx1250) — compile-verified
//
#include <hip/hip_runtime.h>
#include <hip/hip_bf16.h>
#include <stdint.h>

typedef __attribute__((ext_vector_type(2))) float v2f;
typedef __attribute__((ext_vector_type(8))) float v8f;

#define T_LEN 8192
#define TPAD  (T_LEN + 8)   // 4-column zero halo each side
#define HALO  4
#define BATCH 2
#define HID   512
#define KTAGS 60
#define KPAD  64
#define START_TAG 58
#define STOP_TAG  59
#define NEGV -10000.0f

// ---------------- zero fill (halo init) ----------------
__global__ void zero_buf(float* __restrict__ p, size_t n) {
    size_t i = (size_t)blockIdx.x * blockDim.x + threadIdx.x;
    size_t stride = (size_t)gridDim.x * blockDim.x;
    for (; i < n; i += stride) p[i] = 0.f;
}

// ---------------- elementwise residual add (halos stay zero) ----------------
__global__ void add_act(const float* __restrict__ a, const float* __restrict__ b,
                        float* __restrict__ o, size_t n) {
    size_t i = (size_t)blockIdx.x * blockDim.x + threadIdx.x;
    size_t stride = (size_t)gridDim.x * blockDim.x;
    for (; i < n; i += stride) o[i] = a[i] + b[i];
}

// ---------------- input transpose: (T,B,3) -> (B,3,TPAD) interior ----------------
__global__ void transpose_in(const float* __restrict__ s, float* __restrict__ x) {
    int i = blockIdx.x * blockDim.x + threadIdx.x;
    const int total = BATCH * 3 * T_LEN;
    if (i >= total) return;
    int t = i % T_LEN;
    int c = (i / T_LEN) % 3;
    int b = i / (3 * T_LEN);
    x[((size_t)(b * 3 + c)) * TPAD + HALO + t] = s[(t * BATCH + b) * 3 + c];
}

// --- weight repack: w(Co,Cin,KT) -> K-pair interleaved Wp[q][co][2], q = kidx/2,
//     kidx = d*Cin + ci, zero-padded to KDP rows. A-fragment = one b64 load. ---
__global__ void repack_w(const float* __restrict__ w, float* __restrict__ wp,
                         int Cin, int KT, int KDP) {
    int i = blockIdx.x * blockDim.x + threadIdx.x;
    int total = KDP * HID;
    if (i >= total) return;
    int kidx = i / HID;
    int co   = i % HID;
    float v = 0.f;
    if (kidx < Cin * KT) {
        int d  = kidx / Cin;
        int ci = kidx % Cin;
        v = w[(co * Cin + ci) * KT + d];
    }
    wp[((size_t)(kidx >> 1) * HID + co) * 2 + (kidx & 1)] = v;
}

// =====================================================================
// Staged conv (Cin % 16 == 0): implicit GEMM, V_WMMA_F32_16X16X4_F32.
// Block = 8 waves, each wave 32co x 64t; block = 256co x 64t.
// Per-ci panel (16 rows x 72 t-cols incl tap halo) DMA'd to LDS ONCE via
// global_load_async_to_lds_b128 (16B-aligned window), reused for all KT taps.
// Double buffered with s_wait_asynccnt. Row stride 72 -> bank (8r+c)%64:
// lane-halves hit disjoint 16-bank groups, conflict-free reads.
// =====================================================================
template<int CIN, int KT, int PAD>
__global__ __launch_bounds__(256) void conv_wmma_s(
    const float* __restrict__ in, const float* __restrict__ wp,
    const float* __restrict__ bias, float* __restrict__ out)
{
    constexpr int NPAN = CIN / 16;           // per-ci panels (all taps share one panel)
    constexpr int BSTR = 72;                 // LDS row stride (floats): 64 body + 8 halo/pad
    __shared__ float panel[2][16 * BSTR];

    const int tid  = threadIdx.x;
    const int lane = tid & 31;
    const int wave = tid >> 5;
    const int lh   = lane >> 4;
    const int ll   = lane & 15;

    const int t0     = blockIdx.x * 64;
    const int cobase = blockIdx.y * 256 + wave * 32;
    const int b      = blockIdx.z;

    const float* xA = in + (size_t)b * CIN * TPAD;

    v8f acc[2][4] = {};

    // stage rows ci0..ci0+15, cols [t0-4, t0+68) -> window start elem (HALO+t0-4) = t0,
    // byte address t0*4 (+ row*32800): both 16B aligned -> b128 async DMA legal.
    auto issue = [&](int p, int bufi) {
        const int ci0 = p * 16;
        const uint32_t lbase = (uint32_t)(uintptr_t)&panel[bufi][0];
        const float* gbase = xA + (size_t)ci0 * TPAD + (HALO + t0 - 4);
        {   // body: 16 rows x 16 quads (cols 0..63)
            int row = tid >> 4, q = tid & 15;
            uint32_t laddr = lbase + (uint32_t)(row * BSTR + q * 4) * 4u;
            const float* g = gbase + (size_t)row * TPAD + q * 4;
            asm volatile("global_load_async_to_lds_b128 %0, %1, off"
                         :: "v"(laddr), "v"((uint64_t)(uintptr_t)g) : "memory");
        }
        {   // tail quads (cols 64..71), duplicated across waves so every wave
            // issues exactly 2 asyncs per panel (uniform ASYNCcnt)
            int row = (tid & 31) >> 1, q = 16 + (tid & 1);
            uint32_t laddr = lbase + (uint32_t)(row * BSTR + q * 4) * 4u;
            const float* g = gbase + (size_t)row * TPAD + q * 4;
            asm volatile("global_load_async_to_lds_b128 %0, %1, off"
                         :: "v"(laddr), "v"((uint64_t)(uintptr_t)g) : "memory");
        }
    };

    issue(0, 0);
    for (int p = 0; p < NPAN; ++p) {
        if (p + 1 < NPAN) {
            issue(p + 1, (p + 1) & 1);
            asm volatile("s_wait_asynccnt 0x2" ::: "memory");   // panel p landed, p+1 in flight
        } else {
            asm volatile("s_wait_asynccnt 0x0" ::: "memory");
        }
        __syncthreads();

        const float* pan = &panel[p & 1][0];
        const int ci0 = p * 16;
        for (int d = 0; d < KT; ++d) {
            const int coloff = 4 + d - PAD;             // tap shift inside staged window
            #pragma unroll
            for (int c = 0; c < 4; ++c) {
                const int q = ((d * CIN + ci0 + c * 4) >> 1) + lh;   // K-pair row
                v2f afr[2];
                #pragma unroll
                for (int m = 0; m < 2; ++m)             // one b64 load per fragment
                    afr[m] = *(const v2f*)&wp[((size_t)q * HID + cobase + m * 16 + ll) * 2];
                #pragma unroll
                for (int f = 0; f < 4; ++f) {
                    const int col = f * 16 + ll + coloff;
                    const int r   = c * 4 + lh * 2;
                    v2f bfr;
                    bfr.x = pan[r * BSTR + col];
                    bfr.y = pan[(r + 1) * BSTR + col];
                    acc[0][f] = __builtin_amdgcn_wmma_f32_16x16x4_f32(
                        false, afr[0], false, bfr, (short)0, acc[0][f], false, false);
                    acc[1][f] = __builtin_amdgcn_wmma_f32_16x16x4_f32(
                        false, afr[1], false, bfr, (short)0, acc[1][f], false, false);
                }
            }
        }
        __syncthreads();
    }

    #pragma unroll
    for (int m = 0; m < 2; ++m) {
        #pragma unroll
        for (int r = 0; r < 8; ++r) {
            const int co = cobase + m * 16 + r + lh * 8;
            const float bv = bias[co];
            #pragma unroll
            for (int f = 0; f < 4; ++f) {
                const int t = t0 + f * 16 + ll;
                float v = acc[m][f][r] + bv;
                out[((size_t)b * HID + co) * TPAD + HALO + t] = v > 0.f ? v : 0.f;
            }
        }
    }
}

// ---------------- conv1 (Cin=3): tiny, direct-load WMMA path ----------------
__global__ __launch_bounds__(256) void conv1_wmma(
    const float* __restrict__ in, const float* __restrict__ wp,   // paired 12x512
    const float* __restrict__ bias, float* __restrict__ out)
{
    const int lane = threadIdx.x & 31;
    const int wave = threadIdx.x >> 5;
    const int lh   = lane >> 4;
    const int ll   = lane & 15;

    const int t0  = blockIdx.x * 64;
    const int co0 = blockIdx.y * 128 + wave * 16;
    const int b   = blockIdx.z;
    const float* xA = in + (size_t)b * 3 * TPAD;

    v8f acc[4] = {};
    for (int k0 = 0; k0 < 12; k0 += 4) {
        const int ka = k0 + lh * 2;
        const int q  = (k0 >> 1) + lh;
        v2f afr = *(const v2f*)&wp[((size_t)q * HID + co0 + ll) * 2];  // zero rows for kidx>=9
        const int d0 = ka / 3,       c0 = ka - d0 * 3;
        const int d1 = (ka + 1) / 3, c1 = (ka + 1) - d1 * 3;
        #pragma unroll
        for (int f = 0; f < 4; ++f) {
            const int t = t0 + f * 16 + ll;
            v2f bfr;   // halo keeps these in-bounds; zero weights kill k>=9 terms
            bfr.x = xA[(size_t)c0 * TPAD + HALO + t + d0 - 1];
            bfr.y = xA[(size_t)c1 * TPAD + HALO + t + d1 - 1];
            acc[f] = __builtin_amdgcn_wmma_f32_16x16x4_f32(
                false, afr, false, bfr, (short)0, acc[f], false, false);
        }
    }
    #pragma unroll
    for (int r = 0; r < 8; ++r) {
        const int co = co0 + r + lh * 8;
        const float bv = bias[co];
        #pragma unroll
        for (int f = 0; f < 4; ++f) {
            const int t = t0 + f * 16 + ll;
            float v = acc[f][r] + bv;
            out[((size_t)b * HID + co) * TPAD + HALO + t] = v > 0.f ? v : 0.f;
        }
    }
}

// ---------------- fc: feats[t,b,k] = fc_b[k] + sum_h c5[b,h,t]*fc_w[k,h] ----------------
__global__ __launch_bounds__(64) void fc_kernel(
    const float* __restrict__ x, const float* __restrict__ fcw,
    const float* __restrict__ fcb, float* __restrict__ feats)
{
    __shared__ float xs[HID];
    const int t = blockIdx.x;
    const int b = blockIdx.y;
    const float* xp = x + (size_t)b * HID * TPAD + HALO + t;
    for (int h = threadIdx.x; h < HID; h += 64) xs[h] = xp[(size_t)h * TPAD];
    __syncthreads();
    const int k = threadIdx.x;
    if (k < KTAGS) {
        float s = fcb[k];
        const float* wr = fcw + (size_t)k * HID;
        #pragma unroll 8
        for (int h = 0; h < HID; ++h) s += wr[h] * xs[h];
        feats[((size_t)t * BATCH + b) * KTAGS + k] = s;
    }
}

// ---------------- Viterbi: single persistent block, tree argmax per step ----------------
__global__ __launch_bounds__(128) void viterbi_kernel(
    const float* __restrict__ feats, const float* __restrict__ trans,
    int* __restrict__ bp, float* __restrict__ out)
{
    __shared__ float tr[KTAGS * KTAGS];
    __shared__ float fv[2][BATCH * KPAD];

    const int tid = threadIdx.x;
    for (int i = tid; i < KTAGS * KTAGS; i += 128) tr[i] = trans[i];

    const int b = tid >> 6;
    const int k = tid & 63;
    fv[0][b * KPAD + k] = (k == START_TAG) ? 0.f : NEGV;
    __syncthreads();

    float trow[KTAGS];
    if (k < KTAGS) {
        #pragma unroll
        for (int i = 0; i < KTAGS; ++i) trow[i] = tr[k * KTAGS + i];
    }

    float fnext = (k < KTAGS) ? feats[(size_t)b * KTAGS + k] : 0.f;

    int p = 0;
    for (int t = 0; t < T_LEN; ++t) {
        const float myfeat = fnext;
        if (t + 1 < T_LEN && k < KTAGS)
            fnext = feats[((size_t)(t + 1) * BATCH + b) * KTAGS + k];  // prefetch

        if (k < KTAGS) {
            const float* fvp = &fv[p][b * KPAD];
            // independent adds, then depth-6 pairwise max tree (ILP-friendly);
            // strict '>' for the right candidate keeps the left entry on ties (not always the first index)
            float vv[KPAD];
            int   ii[KPAD];
            #pragma unroll
            for (int prev = 0; prev < KTAGS; ++prev) {
                vv[prev] = fvp[prev] + trow[prev];
                ii[prev] = prev;
            }
            #pragma unroll
            for (int prev = KTAGS; prev < KPAD; ++prev) { vv[prev] = -3.4e38f; ii[prev] = 0; }
            #pragma unroll
            for (int s = KPAD / 2; s >= 1; s >>= 1) {
                #pragma unroll
                for (int i = 0; i < s; ++i) {
                    bool gt = vv[i + s] > vv[i];
                    vv[i] = gt ? vv[i + s] : vv[i];
                    ii[i] = gt ? ii[i + s] : ii[i];
                }
            }
            fv[p ^ 1][b * KPAD + k] = vv[0] + myfeat;
            bp[((size_t)t * BATCH + b) * KPAD + k] = ii[0];
        } else {
            fv[p ^ 1][b * KPAD + k] = NEGV;
        }
        p ^= 1;
        __syncthreads();
    }

    if (tid < BATCH) {
        const int bb = tid;
        float best = -3.4e38f;
        int bi = 0;
        for (int kk = 0; kk < KTAGS; ++kk) {
            float v = fv[p][bb * KPAD + kk] + tr[STOP_TAG * KTAGS + kk];
            if (v > best) { best = v; bi = kk; }
        }
        out[bb] = best;
        int tag = bi;
        for (int t = T_LEN - 1; t >= 0; --t) {
            out[2 + (size_t)t * BATCH + bb] = (float)tag;
            tag = bp[((size_t)t * BATCH + bb) * KPAD + tag];
        }
    }
}

extern "C" void kernel_launch(void* const* d_in, const int* in_sizes, int n_in,
                              void* d_out, int out_size, void* d_ws, size_t ws_size,
                              hipStream_t stream) {
    const float* sentence = (const float*)d_in[0];
    const float* w1 = (const float*)d_in[1];  const float* b1 = (const float*)d_in[2];
    const float* w2 = (const float*)d_in[3];  const float* b2 = (const float*)d_in[4];
    const float* w3 = (const float*)d_in[5];  const float* b3 = (const float*)d_in[6];
    const float* w4 = (const float*)d_in[7];  const float* b4 = (const float*)d_in[8];
    const float* w5 = (const float*)d_in[9];  const float* b5 = (const float*)d_in[10];
    const float* fcw = (const float*)d_in[11];
    const float* fcb = (const float*)d_in[12];
    const float* trans = (const float*)d_in[13];

    char* ws = (char*)d_ws;
    size_t cur = 0;
    auto carve = [&](size_t bytes) -> void* {
        void* p = ws + cur;
        cur = (cur + bytes + 255) & ~(size_t)255;
        return p;
    };

    const size_t nAct = (size_t)BATCH * HID * TPAD;
    const size_t nXT  = (size_t)BATCH * 3 * TPAD;
    float* xT   = (float*)carve(nXT * sizeof(float));
    float* bufA = (float*)carve(nAct * sizeof(float));
    float* bufB = (float*)carve(nAct * sizeof(float));
    float* bufC = (float*)carve(nAct * sizeof(float));
    float* wp1  = (float*)carve((size_t)12   * HID * sizeof(float));
    float* wp2  = (float*)carve((size_t)1536 * HID * sizeof(float));
    float* wp3  = (float*)carve((size_t)1536 * HID * sizeof(float));
    float* wp4  = (float*)carve((size_t)1536 * HID * sizeof(float));
    float* wp5  = (float*)carve((size_t)2560 * HID * sizeof(float));
    float* feats = (float*)carve((size_t)T_LEN * BATCH * KTAGS * sizeof(float));
    int*   bp    = (int*)  carve((size_t)T_LEN * BATCH * KPAD * sizeof(int));
    (void)ws_size; (void)in_sizes; (void)n_in; (void)out_size;

    zero_buf<<<1024, 256, 0, stream>>>(xT,   nXT);
    zero_buf<<<4096, 256, 0, stream>>>(bufA, nAct);
    zero_buf<<<4096, 256, 0, stream>>>(bufB, nAct);
    zero_buf<<<4096, 256, 0, stream>>>(bufC, nAct);

    transpose_in<<<(BATCH * 3 * T_LEN + 255) / 256, 256, 0, stream>>>(sentence, xT);
    repack_w<<<(12   * HID + 255) / 256, 256, 0, stream>>>(w1, wp1, 3,   3, 12);
    repack_w<<<(1536 * HID + 255) / 256, 256, 0, stream>>>(w2, wp2, HID, 3, 1536);
    repack_w<<<(1536 * HID + 255) / 256, 256, 0, stream>>>(w3, wp3, HID, 3, 1536);
    repack_w<<<(1536 * HID + 255) / 256, 256, 0, stream>>>(w4, wp4, HID, 3, 1536);
    repack_w<<<(2560 * HID + 255) / 256, 256, 0, stream>>>(w5, wp5, HID, 5, 2560);

    const dim3 cg1(T_LEN / 64, HID / 128, BATCH);
    const dim3 cgs(T_LEN / 64, HID / 256, BATCH);
    const size_t addBlocks = 4096;

    conv1_wmma<<<cg1, 256, 0, stream>>>(xT, wp1, b1, bufA);                  // a2 -> A
    conv_wmma_s<HID, 3, 1><<<cgs, 256, 0, stream>>>(bufA, wp2, b2, bufB);    // a3 -> B
    add_act<<<addBlocks, 256, 0, stream>>>(bufA, bufB, bufC, nAct);          // s23 -> C
    conv_wmma_s<HID, 3, 1><<<cgs, 256, 0, stream>>>(bufC, wp3, b3, bufA);    // a4 -> A
    add_act<<<addBlocks, 256, 0, stream>>>(bufB, bufA, bufC, nAct);          // s34 -> C
    conv_wmma_s<HID, 3, 1><<<cgs, 256, 0, stream>>>(bufC, wp4, b4, bufB);    // a5 -> B
    add_act<<<addBlocks, 256, 0, stream>>>(bufA, bufB, bufC, nAct);          // s45 -> C
    conv_wmma_s<HID, 5, 2><<<cgs, 256, 0, stream>>>(bufC, wp5, b5, bufA);    // c5 -> A

    fc_kernel<<<dim3(T_LEN, BATCH), 64, 0, stream>>>(bufA, fcw, fcb, feats);
    viterbi_kernel<<<1, 128, 0, stream>>>(feats, trans, bp, (float*)d_out);
}